// QCausalConv1D_40243843564242
// MI455X (gfx1250) — compile-verified
//
#include <hip/hip_runtime.h>
#include <hip/hip_bf16.h>

// int8 depthwise causal conv1d (K=4) + bias + SiLU + requant.
// Harness delivers int8 tensors widened to int32 (const int*), scales as
// float*, and wants the int8 result written as float* (one float per elem).
//
// Memory-bound streaming kernel: ~1 GiB of traffic, ~46us floor at 23.3 TB/s.
// Data path: global -> LDS via GLOBAL_LOAD_ASYNC_TO_LDS_B128 (ASYNCcnt, NT),
// halo read from LDS, non-temporal output stores.

#define LROW 4096      // L (row length, elements)
#define TPB  256       // threads per block (8 wave32)
#define EPT  16        // elements per thread: 256*16 == 4096

typedef float v4f __attribute__((ext_vector_type(4)));

__device__ __forceinline__ unsigned lds_off_u32(const void* p) {
    // generic -> LDS(3) addrspace cast; AS3 pointers are the 32-bit LDS offset
    return (unsigned)(unsigned long long)(__attribute__((address_space(3))) const void*)p;
}

__global__ __launch_bounds__(TPB) void qcausal_conv1d_silu_kernel(
    const int* __restrict__ x,       // (B*D, L) int8 values widened to int
    const int* __restrict__ w,       // (D, 4)
    const int* __restrict__ bias,    // (D,)
    const float* __restrict__ in_s,
    const float* __restrict__ w_s,
    const float* __restrict__ out_s,
    const float* __restrict__ b_s,
    float* __restrict__ out,         // (B*D, L) int8 result as float
    int Dmask)                       // D-1 (D is a power of two: 4096)
{
    // x[l] lives at smem[l + 4]; smem[0..3] = 0 gives the causal left pad.
    __shared__ int smem[LROW + 4 + 12];  // +12 pad (keeps size 16B-multiple)

    const int t   = threadIdx.x;
    const int row = blockIdx.x;
    const int d   = row & Dmask;

    if (t < 4) smem[t] = 0;

    const long long rowbase = (long long)row * LROW;
    const int* gsrc = x + rowbase;

    // Stream the whole row into LDS: 1024 x 16B async copies, 4 per lane.
    // Lane-contiguous 16B chunks per instruction -> fully coalesced.
    // NT hint: input is read exactly once, keep it out of L2 residency.
#pragma unroll
    for (int i = 0; i < 4; ++i) {
        const int c = t + TPB * i;                       // 16-byte chunk id
        unsigned   lb = lds_off_u32(&smem[4 + c * 4]);   // 16B-aligned LDS dest
        const int* gp = gsrc + c * 4;                    // 16B-aligned source
        asm volatile("global_load_async_to_lds_b128 %0, %1, off th:TH_LOAD_NT"
                     :: "v"(lb), "v"(gp) : "memory");
    }
    asm volatile("s_wait_asynccnt 0" ::: "memory");
    __syncthreads();

    // Per-channel taps and scales (uniform per block; cheap, L2-resident).
    const int4  wt  = *(const int4*)(w + d * 4);
    const float sxw = in_s[0] * w_s[0];                  // fold both dequants
    const float bf  = (float)bias[d] * b_s[0];
    const float inv_out = 1.0f / out_s[0];

    // This thread's 16 outputs need x[l0-3 .. l0+15] -> smem[l0 .. l0+19].
    const int l0 = t * EPT;
    int xv[20];
    const int4* sp = (const int4*)&smem[l0];             // 16B-aligned
#pragma unroll
    for (int i = 0; i < 5; ++i) {
        int4 q = sp[i];                                   // ds_load_b128
        xv[i * 4 + 0] = q.x; xv[i * 4 + 1] = q.y;
        xv[i * 4 + 2] = q.z; xv[i * 4 + 3] = q.w;
    }

    float* orow = out + rowbase;
#pragma unroll
    for (int g = 0; g < 4; ++g) {
        v4f o;
#pragma unroll
        for (int e4 = 0; e4 < 4; ++e4) {
            const int e = g * 4 + e4;
            // integer 4-tap dot (exact), then one scale to float
            const int idot = wt.x * xv[e + 1] + wt.y * xv[e + 2]
                           + wt.z * xv[e + 3] + wt.w * xv[e + 4];
            const float y = (float)idot * sxw + bf;
            // silu: y * sigmoid(y)  (v_exp_f32 + v_rcp_f32)
            const float s = y * __builtin_amdgcn_rcpf(1.0f + __expf(-y));
            // requant: round-half-even (v_rndne) + clamp to int8 range
            float q = __builtin_rintf(s * inv_out);
            q = fminf(fmaxf(q, -128.0f), 127.0f);
            o[e4] = q;
        }
        __builtin_nontemporal_store(o, (v4f*)(orow + l0 + g * 4));
    }
}

extern "C" void kernel_launch(void* const* d_in, const int* in_sizes, int n_in,
                              void* d_out, int out_size, void* d_ws, size_t ws_size,
                              hipStream_t stream) {
    const int*   x    = (const int*)d_in[0];     // (B, D, L) int8 -> int
    const int*   w    = (const int*)d_in[1];     // (D, K=4)
    const int*   bias = (const int*)d_in[2];     // (D,)
    const float* is   = (const float*)d_in[3];
    const float* ws   = (const float*)d_in[4];
    const float* os   = (const float*)d_in[5];
    const float* bs   = (const float*)d_in[6];
    float*       out  = (float*)d_out;

    const int D    = in_sizes[2];                // 4096 (power of two)
    const int rows = in_sizes[0] / LROW;         // B*D = 32768

    qcausal_conv1d_silu_kernel<<<dim3(rows), dim3(TPB), 0, stream>>>(
        x, w, bias, is, ws, os, bs, out, D - 1);
}